// DiagLRConv_34643206209663
// MI455X (gfx1250) — compile-verified
//
#include <hip/hip_runtime.h>

// out[n,o,h,w] = sum_k sum_i W[o,i,k] * x[n,i,h+k-2,w+k-2]  (5 diagonal taps, pad=2)
// One wave32 computes a [16 out-ch x 16 w] f32 tile via 20x V_WMMA_F32_16X16X4_F32.

typedef __attribute__((ext_vector_type(2))) float v2f;
typedef __attribute__((ext_vector_type(8))) float v8f;

#define C_IN   16
#define C_OUT  16
#define HW     512
#define KTAPS  5
#define PAD    2
#define WTILES (HW / 16)            // 32 w-tiles per row
#define WAVES_PER_BLOCK 8           // 256 threads = 8 wave32

__global__ __launch_bounds__(256) void diag_conv_wmma_f32(
    const float* __restrict__ x,    // [16,16,512,512] NCHW
    const float* __restrict__ wflt, // [16,16,5]  (o,i,k)
    float* __restrict__ out)        // [16,16,512,512]
{
  const int lane = threadIdx.x & 31;
  const int wib  = threadIdx.x >> 5;          // wave in block
  const int half = lane >> 4;                 // 0: lanes 0-15, 1: lanes 16-31
  const int idx  = lane & 15;

  // Tile id ordered so h is fastest-varying: the 8 waves of a block cover
  // 8 consecutive output rows of the same (n, w-strip) -> input rows are
  // shared 5x between neighboring waves and hit in WGP$/L2.
  long tile = (long)blockIdx.x * WAVES_PER_BLOCK + wib;
  const int h   = (int)(tile % HW);
  long t2       = tile / HW;
  const int wt  = (int)(t2 % WTILES);
  const int n   = (int)(t2 / WTILES);
  const int w0  = wt * 16;

  // ---- A matrices: W per tap, split into 4 chunks of K=4 in-channels ----
  // f32 16x4 A layout: lanes 0-15 hold M=lane with K={0,1} in v0/v1,
  // lanes 16-31 hold M=lane-16 with K={2,3}.  => this lane: M=idx, K=2*half+{0,1}
  v2f a[KTAPS][4];
#pragma unroll
  for (int k = 0; k < KTAPS; ++k) {
#pragma unroll
    for (int c = 0; c < 4; ++c) {
      const int i0 = 4 * c + 2 * half;
      a[k][c].x = wflt[(idx * C_IN + i0    ) * KTAPS + k];
      a[k][c].y = wflt[(idx * C_IN + i0 + 1) * KTAPS + k];
    }
  }

  v8f acc = {};
  const float* xn = x + (long)n * C_IN * HW * HW;

#pragma unroll
  for (int k = 0; k < KTAPS; ++k) {
    const int row = h + k - PAD;
    if (row < 0 || row >= HW) continue;       // wave-uniform: EXEC stays all-1s

    // B is K(4 in-ch) x N(16 w).  Lane: N=idx, K=2*half+{0,1}.
    const int col   = w0 + idx + k - PAD;
    const bool cOk  = (col >= 0) && (col < HW);
    const int  colc = col < 0 ? 0 : (col >= HW ? HW - 1 : col);   // clamp addr
    const long base = (long)row * HW + colc;

#pragma unroll
    for (int c = 0; c < 4; ++c) {
      const int i0 = 4 * c + 2 * half;
      float b0 = xn[(long)(i0    ) * HW * HW + base];
      float b1 = xn[(long)(i0 + 1) * HW * HW + base];
      v2f b;
      b.x = cOk ? b0 : 0.0f;                  // v_cndmask, loads stay coalesced
      b.y = cOk ? b1 : 0.0f;
      // args: (neg_a, A, neg_b, B, c_mod, C, reuse_a, reuse_b)
      acc = __builtin_amdgcn_wmma_f32_16x16x4_f32(
          false, a[k][c], false, b, (short)0, acc, false, false);
    }
  }

  // ---- Store D: VGPR r <-> out-channel o = r + 8*half, N = idx ----
  float* on = out + (((long)n * C_OUT) * HW + h) * HW + w0 + idx;
#pragma unroll
  for (int r = 0; r < 8; ++r) {
    const int o = r + 8 * half;
    on[(long)o * HW * HW] = acc[r];
  }
}

extern "C" void kernel_launch(void* const* d_in, const int* in_sizes, int n_in,
                              void* d_out, int out_size, void* d_ws, size_t ws_size,
                              hipStream_t stream) {
  const float* x    = (const float*)d_in[0];   // [16,16,512,512]
  const float* wflt = (const float*)d_in[1];   // [16,16,5]
  float* out        = (float*)d_out;           // [16,16,512,512]

  const long tiles  = 16L * HW * WTILES;       // n * h * w-tiles = 262144
  const int  blocks = (int)(tiles / WAVES_PER_BLOCK);   // 32768

  diag_conv_wmma_f32<<<blocks, 256, 0, stream>>>(x, wflt, out);
}